// GCNConv1_21818433863944
// MI455X (gfx1250) — compile-verified
//
#include <hip/hip_runtime.h>

typedef __attribute__((ext_vector_type(2))) float v2f;
typedef __attribute__((ext_vector_type(8))) float v8f;

#define FDIM 128
#define LDSP (FDIM + 4)   // padded LDS stride (132) -> conflict-free b64 reads

// ---------------- degree / norm prep ----------------
__global__ void k_fill1(float* __restrict__ p, int n) {
    int i = blockIdx.x * blockDim.x + threadIdx.x;
    if (i < n) p[i] = 1.0f;                       // self-loop weight
}

__global__ void k_deg_edges(const int* __restrict__ dst, const float* __restrict__ ew,
                            float* __restrict__ deg, int E) {
    int e = blockIdx.x * blockDim.x + threadIdx.x;
    if (e < E) atomicAdd(&deg[dst[e]], ew[e]);
}

__global__ void k_rsqrt(float* __restrict__ p, int n) {
    int i = blockIdx.x * blockDim.x + threadIdx.x;
    if (i < n) p[i] = rsqrtf(p[i]);               // deg >= 1 always (self loop)
}

// ------- fp32 WMMA GEMM: Out[N,128] = X[N,128] @ W[128,128] -------
// block = 256 threads = 8 waves; block -> 32-row tile (two 16-row WMMA tiles,
// two independent accumulator chains per wave); wave w -> 16-col tile.
// W is staged TRANSPOSED in LDS so each B fragment is one aligned ds_load_b64.
__global__ __launch_bounds__(256) void k_gemm_wmma(const float* __restrict__ X,
                                                   const float* __restrict__ W,
                                                   float* __restrict__ Out, int N) {
    __shared__ float sWt[FDIM * LDSP];  // W^T: sWt[n*LDSP + k] = W[k,n]  (67.6 KB)
    __shared__ float sX[32 * LDSP];     // 32 input rows                  (16.9 KB)

    int tid = threadIdx.x;
    for (int i = tid; i < FDIM * FDIM; i += 256) {
        int k = i >> 7, n = i & 127;                 // coalesced global read
        sWt[n * LDSP + k] = W[i];
    }
    int row0 = blockIdx.x * 32;
    for (int i = tid; i < 32 * FDIM; i += 256) {
        int r = i >> 7, c = i & 127;
        int row = row0 + r; if (row >= N) row = N - 1;
        sX[r * LDSP + c] = X[(size_t)row * FDIM + c];
    }
    __syncthreads();

    int wave = tid >> 5;
    int lane = tid & 31;
    int m    = lane & 15;            // A row within tile / B,C column index
    int khi  = (lane >> 4) << 1;     // K sub-offset: 0 (lanes 0-15) or 2 (lanes 16-31)
    int col0 = wave << 4;

    v8f acc0 = {};
    v8f acc1 = {};
    #pragma unroll
    for (int k = 0; k < FDIM; k += 4) {
        // A 16x4 f32 fragments for the two row tiles (aligned float2 LDS loads)
        v2f a0 = *(const v2f*)&sX[m * LDSP + k + khi];
        v2f a1 = *(const v2f*)&sX[(16 + m) * LDSP + k + khi];
        // B 4x16 f32 fragment from transposed W: rows k+khi, k+khi+1 of column col0+m
        v2f b = *(const v2f*)&sWt[(col0 + m) * LDSP + k + khi];
        acc0 = __builtin_amdgcn_wmma_f32_16x16x4_f32(false, a0, false, b,
                                                     (short)0, acc0, false, false);
        acc1 = __builtin_amdgcn_wmma_f32_16x16x4_f32(false, a1, false, b,
                                                     (short)0, acc1, false, false);
    }

    // C/D layout: VGPR r -> M = r (lanes 0-15) / M = 8+r (lanes 16-31), N = lane&15
    int mb = (lane >> 4) << 3;
    #pragma unroll
    for (int r = 0; r < 8; ++r) {
        int row = row0 + mb + r;
        if (row < N) Out[(size_t)row * FDIM + col0 + m] = acc0[r];
        if (row + 16 < N) Out[(size_t)(row + 16) * FDIM + col0 + m] = acc1[r];
    }
}

// ---------------- aggregation ----------------
// out[i] = h[i] * dis[i]^2  (self-loop contribution, also initializes out)
__global__ void k_self(const float* __restrict__ h, const float* __restrict__ dis,
                       float* __restrict__ out, int total) {
    int idx = blockIdx.x * blockDim.x + threadIdx.x;
    if (idx < total) {
        float d = dis[idx >> 7];
        out[idx] = h[idx] * d * d;
    }
}

// one wave32 per edge; lane handles 4 features (float4 gather + 4 f32 L2 atomics)
__global__ void k_scatter(const int* __restrict__ src, const int* __restrict__ dst,
                          const float* __restrict__ ew, const float* __restrict__ dis,
                          const float* __restrict__ h, float* __restrict__ out, int E) {
    int wid  = (blockIdx.x * blockDim.x + threadIdx.x) >> 5;
    int lane = threadIdx.x & 31;
    if (wid >= E) return;
    int s = src[wid], d = dst[wid];
    float c = dis[s] * ew[wid] * dis[d];
    float4 v = ((const float4*)(h + (size_t)s * FDIM))[lane];
    float* o = out + (size_t)d * FDIM + lane * 4;
    atomicAdd(o + 0, v.x * c);
    atomicAdd(o + 1, v.y * c);
    atomicAdd(o + 2, v.z * c);
    atomicAdd(o + 3, v.w * c);
}

__global__ void k_bias_relu(float* __restrict__ p, const float* __restrict__ b, int total) {
    int idx = blockIdx.x * blockDim.x + threadIdx.x;
    if (idx < total) {
        float v = p[idx] + b[idx & 127];
        p[idx] = v > 0.f ? v : 0.f;
    }
}

// ---------------- readout ----------------
__global__ void k_zero(float* __restrict__ p, int n) {
    int i = blockIdx.x * blockDim.x + threadIdx.x;
    if (i < n) p[i] = 0.f;
}

// block = 128 threads (one per feature, coalesced row reads), 256 nodes per block
__global__ __launch_bounds__(128) void k_colsum(const float* __restrict__ h,
                                                float* __restrict__ gsum, int N) {
    int f  = threadIdx.x;
    int i0 = blockIdx.x * 256;
    int i1 = i0 + 256; if (i1 > N) i1 = N;
    float s = 0.f;
    for (int i = i0; i < i1; ++i) s += h[(size_t)i * FDIM + f];
    atomicAdd(&gsum[f], s);
}

__global__ __launch_bounds__(128) void k_mlp(const float* __restrict__ gsum,
                                             const float* __restrict__ lw1,
                                             const float* __restrict__ lb1,
                                             const float* __restrict__ lw2,
                                             const float* __restrict__ lb2,
                                             float* __restrict__ out, int N, int H) {
    __shared__ float g[FDIM];
    __shared__ float t[FDIM];
    int tid = threadIdx.x;
    g[tid] = gsum[tid] / (float)N;
    __syncthreads();
    float acc = 0.f;
    if (tid < H) {
        for (int f = 0; f < FDIM; ++f) acc += g[f] * lw1[f * H + tid];
        acc += lb1[tid];
        acc = acc > 0.f ? acc : 0.f;
        acc *= lw2[tid];
    }
    t[tid] = acc;   // zeros for tid >= H
    __syncthreads();
    for (int s = 64; s > 0; s >>= 1) {
        if (tid < s) t[tid] += t[tid + s];
        __syncthreads();
    }
    if (tid == 0) out[0] = t[0] + lb2[0];
}

// ---------------- driver ----------------
extern "C" void kernel_launch(void* const* d_in, const int* in_sizes, int n_in,
                              void* d_out, int out_size, void* d_ws, size_t ws_size,
                              hipStream_t stream) {
    const float* x   = (const float*)d_in[0];
    const int*   ei  = (const int*)d_in[1];
    const float* ew  = (const float*)d_in[2];
    const float* W1  = (const float*)d_in[3];
    const float* b1  = (const float*)d_in[4];
    const float* W2  = (const float*)d_in[5];
    const float* b2  = (const float*)d_in[6];
    const float* lw1 = (const float*)d_in[7];
    const float* lb1 = (const float*)d_in[8];
    const float* lw2 = (const float*)d_in[9];
    const float* lb2 = (const float*)d_in[10];
    float* out = (float*)d_out;

    int N = in_sizes[0] / FDIM;
    int E = in_sizes[2];
    int H = in_sizes[8];
    const int* src = ei;
    const int* dst = ei + E;

    float* ws   = (float*)d_ws;
    float* dis  = ws;                                   // N floats
    float* A    = ws + (size_t)((N + 255) & ~255);      // N*128 (GEMM out)
    float* B    = A + (size_t)N * FDIM;                 // N*128 (agg out)
    float* gsum = B + (size_t)N * FDIM;                 // 128

    int total = N * FDIM;
    dim3 blk(256);
    int gN    = (N + 255) / 256;
    int gE    = (E + 255) / 256;
    int gEw   = (E + 7) / 8;          // 8 waves/block, one wave per edge
    int gT    = (total + 255) / 256;
    int gTile = (N + 31) / 32;        // 32 rows per block

    // symmetric degree normalization
    k_fill1    <<<gN, blk, 0, stream>>>(dis, N);
    k_deg_edges<<<gE, blk, 0, stream>>>(dst, ew, dis, E);
    k_rsqrt    <<<gN, blk, 0, stream>>>(dis, N);

    // layer 1
    k_gemm_wmma<<<gTile, blk, 0, stream>>>(x, W1, A, N);
    k_self     <<<gT,   blk, 0, stream>>>(A, dis, B, total);
    k_scatter  <<<gEw,  blk, 0, stream>>>(src, dst, ew, dis, A, B, E);
    k_bias_relu<<<gT,   blk, 0, stream>>>(B, b1, total);

    // layer 2 (ping-pong A/B)
    k_gemm_wmma<<<gTile, blk, 0, stream>>>(B, W2, A, N);
    k_self     <<<gT,   blk, 0, stream>>>(A, dis, B, total);
    k_scatter  <<<gEw,  blk, 0, stream>>>(src, dst, ew, dis, A, B, E);
    k_bias_relu<<<gT,   blk, 0, stream>>>(B, b2, total);

    // mean readout + MLP head
    k_zero  <<<1, 128, 0, stream>>>(gsum, FDIM);
    k_colsum<<<(N + 255) / 256, 128, 0, stream>>>(B, gsum, N);
    k_mlp   <<<1, 128, 0, stream>>>(gsum, lw1, lb1, lw2, lb2, out, N, H);
}